// GraphSAGELayer_41875931136731
// MI455X (gfx1250) — compile-verified
//
#include <hip/hip_runtime.h>

#define N_NODES 100000
#define N_EDGES 1600000
#define F 128            // IN_FEATS == OUT_FEATS
#define K2 256           // 2*F, GEMM reduction dim
#define ROWS_PER_BLOCK 32
#define LDS_STRIDE 260   // 256 + 4 pad -> bank-conflict-free column access

typedef __attribute__((ext_vector_type(2))) float v2f;
typedef __attribute__((ext_vector_type(8))) float v8f;

// ---------------------------------------------------------------- zero aggr
__global__ void sage_zero(float* __restrict__ p, long n) {
    long i = (long)blockIdx.x * blockDim.x + threadIdx.x;
    long stride = (long)gridDim.x * blockDim.x;
    for (; i < n; i += stride) p[i] = 0.0f;
}

// ------------------------------------------------- edge scatter (wave/edge)
// wave32: one wave per edge -> adj_{rows,cols,vals} are wave-uniform (scalar
// loads), each lane moves one float4 of the 128-float feature row (512B
// coalesced), then 4 hardware f32 atomics into aggr (L2-resident, 51MB<192MB).
__global__ __launch_bounds__(256) void sage_scatter(
    const float* __restrict__ x, const int* __restrict__ rows,
    const int* __restrict__ cols, const float* __restrict__ vals,
    float* __restrict__ aggr) {
    unsigned gid  = blockIdx.x * 256u + threadIdx.x;
    unsigned e    = gid >> 5;
    unsigned lane = gid & 31u;
    if (e >= N_EDGES) return;
    int   r = rows[e];
    int   c = cols[e];
    float v = vals[e];
    float4 xv = ((const float4*)(x + (size_t)c * F))[lane];
    float* dst = aggr + (size_t)r * F + lane * 4u;
    unsafeAtomicAdd(dst + 0, v * xv.x);
    unsafeAtomicAdd(dst + 1, v * xv.y);
    unsafeAtomicAdd(dst + 2, v * xv.z);
    unsafeAtomicAdd(dst + 3, v * xv.w);
}

// --------------------------------------------- fused concat + GEMM + ReLU
// Block = 8 waves = 32 rows x 128 cols of output. A-panel [32 x 256] staged
// in LDS once; wave w owns output columns [16w, 16w+16) and accumulates two
// 16x16 tiles with V_WMMA_F32_16X16X4_F32 (exact f32 math).
__global__ __launch_bounds__(256) void sage_gemm(
    const float* __restrict__ x, const float* __restrict__ aggr,
    const float* __restrict__ w, float* __restrict__ out) {
    __shared__ float lds[ROWS_PER_BLOCK * LDS_STRIDE];

    const unsigned t  = threadIdx.x;
    const unsigned m0 = blockIdx.x * ROWS_PER_BLOCK;   // 100000 % 32 == 0

    // Stage combined = [x | aggr] rows m0..m0+31 into LDS (float4 granules).
    #pragma unroll
    for (int i = 0; i < 8; ++i) {
        unsigned idx = t + 256u * i;      // float4 index; 64 float4 per row
        unsigned row = idx >> 6;
        unsigned col = (idx & 63u) * 4u;
        const float* src = (col < F)
            ? (x    + (size_t)(m0 + row) * F + col)
            : (aggr + (size_t)(m0 + row) * F + (col - F));
        float4 val = *(const float4*)src;
        float* d = &lds[row * LDS_STRIDE + col];
        d[0] = val.x; d[1] = val.y; d[2] = val.z; d[3] = val.w;
    }
    __syncthreads();

    const unsigned wave = t >> 5;
    const unsigned lane = t & 31u;
    const unsigned half = lane >> 4;      // ISA: lanes>=16 hold K+2,K+3
    const unsigned l16  = lane & 15u;     // M index (A) / N index (B)
    const unsigned n0   = wave * 16u;     // this wave's output column tile

    v8f acc0 = {};
    v8f acc1 = {};
    const float* a0p = &lds[l16 * LDS_STRIDE];          // rows m0..m0+15
    const float* a1p = &lds[(l16 + 16u) * LDS_STRIDE];  // rows m0+16..m0+31
    const float* bp  = w + n0 + l16;                    // W is [256 x 128] row-major

    #pragma unroll 4
    for (int kk = 0; kk < K2; kk += 4) {
        int ko = kk + 2 * (int)half;
        v2f a0 = *(const v2f*)(a0p + ko);   // A[l16][ko], A[l16][ko+1]
        v2f a1 = *(const v2f*)(a1p + ko);
        v2f b;
        b.x = bp[(size_t)ko * F];           // B[ko][n0+l16]
        b.y = bp[(size_t)(ko + 1) * F];     // B[ko+1][n0+l16]
        acc0 = __builtin_amdgcn_wmma_f32_16x16x4_f32(
                   false, a0, false, b, (short)0, acc0, false, false);
        acc1 = __builtin_amdgcn_wmma_f32_16x16x4_f32(
                   false, a1, false, b, (short)0, acc1, false, false);
    }

    // C/D layout: VGPR j holds M = j + 8*half, N = n0 + l16. Fused ReLU.
    const unsigned colg = n0 + l16;
    #pragma unroll
    for (int j = 0; j < 8; ++j) {
        unsigned r0 = m0 + 8u * half + (unsigned)j;   // tile 0
        unsigned r1 = r0 + 16u;                       // tile 1
        out[(size_t)r0 * F + colg] = fmaxf(acc0[j], 0.0f);
        out[(size_t)r1 * F + colg] = fmaxf(acc1[j], 0.0f);
    }
}

// ---------------------------------------------------------------- launcher
extern "C" void kernel_launch(void* const* d_in, const int* in_sizes, int n_in,
                              void* d_out, int out_size, void* d_ws, size_t ws_size,
                              hipStream_t stream) {
    const float* x    = (const float*)d_in[0];
    const int*   rows = (const int*)  d_in[1];
    const int*   cols = (const int*)  d_in[2];
    const float* vals = (const float*)d_in[3];
    const float* wgt  = (const float*)d_in[4];
    float* out  = (float*)d_out;
    float* aggr = (float*)d_ws;     // [N_NODES * F] f32 scratch (51.2 MB)

    (void)in_sizes; (void)n_in; (void)out_size; (void)ws_size;

    sage_zero<<<2048, 256, 0, stream>>>(aggr, (long)N_NODES * F);

    // one wave32 per edge -> E*32 threads
    unsigned scatter_blocks = (N_EDGES * 32u + 255u) / 256u;   // 200000
    sage_scatter<<<scatter_blocks, 256, 0, stream>>>(x, rows, cols, vals, aggr);

    sage_gemm<<<N_NODES / ROWS_PER_BLOCK, 256, 0, stream>>>(x, aggr, wgt, out);
}